// SpVecsUnet_38869454029179
// MI455X (gfx1250) — compile-verified
//
#include <hip/hip_runtime.h>
#include <math.h>

// ---------------------------------------------------------------------------
// SpVecsUnet superpixel mean pooling for MI455X (gfx1250, wave32).
//
//   features  : [C=64, H, W] f32      (d_in[0], C*H*W elements)
//   pix_yx    : [N, 2] i32            (d_in[1])
//   segment_ids: [N] i32              (d_in[2])
//   n_segments : scalar i32           (d_in[3], device-side; S derived from out_size)
//   out       : [S, C] f32 = segment mean
//
// Memory-bound: 64M scattered 4B gathers through L2 (feature map ~256MB,
// L2=192MB, each 64B line reused ~16x), plus 64M fp32 atomic adds into a
// 1MB L2-resident accumulator. Final normalization done as a block-diagonal
// f32 matmul on the WMMA units (V_WMMA_F32_16X16X4_F32).
// ---------------------------------------------------------------------------

typedef float v2f __attribute__((ext_vector_type(2)));
typedef float v8f __attribute__((ext_vector_type(8)));

#define C_CH 64

__device__ __forceinline__ void atomic_fadd(float* p, float v) {
#if defined(__gfx1250__)
  // native global_atomic_add_f32 (non-returning -> STOREcnt, executed at L2)
  unsafeAtomicAdd(p, v);
#else
  atomicAdd(p, v);
#endif
}

// --- Kernel 0: zero the accumulators (sums in d_out, counts in d_ws) -------
__global__ void SpVecs_zero(float* __restrict__ sums, float* __restrict__ cnts,
                            int nsum, int ncnt) {
  int i = blockIdx.x * blockDim.x + threadIdx.x;
  if (i < nsum) sums[i] = 0.0f;
  if (i < ncnt) cnts[i] = 0.0f;
}

// --- Kernel 1: gather 64 channels per pixel, scatter-add into L2 -----------
__global__ __launch_bounds__(256) void SpVecs_scatter(
    const float* __restrict__ feat, const int* __restrict__ yx,
    const int* __restrict__ seg, float* __restrict__ sums,
    float* __restrict__ cnts, int N, int HW, int W) {
  int i = blockIdx.x * blockDim.x + threadIdx.x;
  if (i >= N) return;

  int y = yx[2 * i];
  int x = yx[2 * i + 1];
  int s = seg[i];

  const float* f = feat + (size_t)y * (size_t)W + (size_t)x;
  float* o = sums + (size_t)s * C_CH;

  atomic_fadd(&cnts[s], 1.0f);

  // Batch 16 strided loads (channel planes are HW floats apart) to build up
  // memory-level parallelism against LOADcnt, then fire the 16 atomics.
  #pragma unroll
  for (int cb = 0; cb < C_CH; cb += 16) {
    float v[16];
    #pragma unroll
    for (int k = 0; k < 16; ++k) v[k] = f[(size_t)(cb + k) * (size_t)HW];
    #pragma unroll
    for (int k = 0; k < 16; ++k) atomic_fadd(&o[cb + k], v[k]);
  }
}

// --- Kernel 2: out[s][c] = sums[s][c] / max(count[s],1) via WMMA -----------
// One wave handles a 16-segment group (rows s0..s0+15, all 64 channels).
// Each 16x16 tile: D = sum_{j=0..3} A_j (16x4, one-hot * 1/count) x B_j
// (4x16 rows of sums), exact f32 via V_WMMA_F32_16X16X4_F32 accumulation.
//
// ISA 7.12.2 layouts (wave32):
//   A 16x4 f32:  lane l: M = l%16; VGPR0 -> K = (l<16 ? 0 : 2), VGPR1 -> K+1
//   B  4x16 f32: lane l: N = l%16; VGPR0 -> K = (l<16 ? 0 : 2), VGPR1 -> K+1
//   D 16x16 f32: VGPR r, lane l: row = r + (l<16 ? 0 : 8), col = l%16
__global__ __launch_bounds__(256) void SpVecs_finalize(
    float* __restrict__ out, const float* __restrict__ cnts) {
  const int lane  = threadIdx.x & 31;
  const int wid   = threadIdx.x >> 5;
  const int g     = blockIdx.x * (blockDim.x >> 5) + wid;  // 16-segment group
  const int s0    = g << 4;
  const int m     = lane & 15;
  const int half  = lane >> 4;        // 0: lanes 0-15, 1: lanes 16-31
  const int khalf = half << 1;        // K base carried by this half (0 or 2)

  const float invc = 1.0f / fmaxf(cnts[s0 + m], 1.0f);

#if __has_builtin(__builtin_amdgcn_wmma_f32_16x16x4_f32)
  #pragma unroll
  for (int c0 = 0; c0 < C_CH; c0 += 16) {
    v8f acc = {0.f, 0.f, 0.f, 0.f, 0.f, 0.f, 0.f, 0.f};
    #pragma unroll
    for (int j = 0; j < 4; ++j) {
      // A_j[m][k] = invc[m] when (4j + k) == m, else 0  (one-hot selector)
      v2f A;
      A.x = (m == 4 * j + khalf + 0) ? invc : 0.0f;
      A.y = (m == 4 * j + khalf + 1) ? invc : 0.0f;
      // B_j[k][n] = sums[s0 + 4j + k][c0 + n]
      const int r0 = s0 + 4 * j + khalf;
      v2f B;
      B.x = out[(size_t)r0 * C_CH + c0 + m];
      B.y = out[(size_t)(r0 + 1) * C_CH + c0 + m];
      // (neg_a, A, neg_b, B, c_mod, C, reuse_a, reuse_b)
      acc = __builtin_amdgcn_wmma_f32_16x16x4_f32(
          false, A, false, B, (short)0, acc, false, false);
    }
    // Write back D (reads above complete before stores via data dependency,
    // and groups/tiles are disjoint, so in-place is safe).
    const int rowbase = s0 + (half << 3);
    #pragma unroll
    for (int r = 0; r < 8; ++r)
      out[(size_t)(rowbase + r) * C_CH + c0 + m] = acc[r];
  }
#else
  // Fallback: plain VALU scale (kept compile-clean if builtin unavailable).
  for (int c0 = 0; c0 < C_CH; c0 += 16) {
    for (int r = half * 8; r < half * 8 + 8; ++r)
      out[(size_t)(s0 + r) * C_CH + c0 + m] *= 1.0f / fmaxf(cnts[s0 + r], 1.0f);
  }
#endif
}

extern "C" void kernel_launch(void* const* d_in, const int* in_sizes, int n_in,
                              void* d_out, int out_size, void* d_ws, size_t ws_size,
                              hipStream_t stream) {
  const float* feat = (const float*)d_in[0];
  const int*   yx   = (const int*)d_in[1];
  const int*   seg  = (const int*)d_in[2];
  float* out  = (float*)d_out;   // doubles as the sums accumulator
  float* cnts = (float*)d_ws;    // S floats (16 KB)

  const int N  = in_sizes[2];            // number of pixel entries
  const int S  = out_size / C_CH;        // number of superpixels
  const int HW = in_sizes[0] / C_CH;     // H*W
  const int W  = (int)lround(sqrt((double)HW));  // square feature map (H==W)

  // 1) zero sums (in d_out) and counts (in d_ws)
  {
    int n = out_size;
    SpVecs_zero<<<(n + 255) / 256, 256, 0, stream>>>(out, cnts, n, S);
  }

  // 2) gather + atomic scatter-add
  SpVecs_scatter<<<(N + 255) / 256, 256, 0, stream>>>(feat, yx, seg, out, cnts,
                                                      N, HW, W);

  // 3) normalize via WMMA; one wave per 16-segment group, 8 waves per block
  {
    int groups = S / 16;                 // e.g. 256 for S=4096
    int wpb = 8;
    int blocks = (groups + wpb - 1) / wpb;
    SpVecs_finalize<<<blocks, wpb * 32, 0, stream>>>(out, cnts);
  }
}